// GCNAggregator_74131135529474
// MI455X (gfx1250) — compile-verified
//
#include <hip/hip_runtime.h>
#include <stdint.h>

// GCN mean-aggregation: out[i] = (sum_{e in seg(i)} F[nbr[e]] + F[i]) / (deg_i + 1)
// Memory-bound gather (feature table is L2-resident); compute negligible -> no WMMA.
// CDNA5 TDM gather-mode (tensor_load_to_lds) DMAs 8 indexed feature rows per op into
// LDS, double buffered on TENSORcnt. Segment bounds come from a precomputed offset
// table (boundary-fill pass) -> one s_load_b64 per wave instead of a ~42-step
// latency-bound scalar binary search.

#define D_FEAT 256
#define ROWS_PER_BATCH 8
#define WAVES_PER_BLOCK 4
#define THREADS_PER_BLOCK (WAVES_PER_BLOCK * 32)

typedef float    vf4  __attribute__((ext_vector_type(4)));
typedef uint32_t su4  __attribute__((ext_vector_type(4)));
typedef uint32_t su8  __attribute__((ext_vector_type(8)));

__device__ __forceinline__ int dmin(int a, int b) { return a < b ? a : b; }

// ---- pass 1: offs[n] = lower_bound(seg, n) for n in [0, N]; offs[N] = E ----
// Edge e writes offs[n] = e for all n in (seg[e-1], seg[e]]  (seg[-1] := -1);
// the last edge also fills (seg[E-1], N] with E. Every n covered exactly once.
__global__ void segment_bounds_kernel(const int* __restrict__ seg,
                                      int* __restrict__ offs, int N, int E)
{
    int e = blockIdx.x * blockDim.x + threadIdx.x;
    if (e >= E) return;
    int s = seg[e];
    int p = (e == 0) ? -1 : seg[e - 1];
    for (int n = p + 1; n <= s; ++n) offs[n] = e;
    if (e == E - 1)
        for (int n = s + 1; n <= N; ++n) offs[n] = E;
}

__global__ __launch_bounds__(THREADS_PER_BLOCK) void
gcn_aggregate_tdm_kernel(const float* __restrict__ feat,
                         const int*   __restrict__ nbr,
                         const int*   __restrict__ seg,
                         const int*   __restrict__ offs,   // may be null -> binary search
                         float*       __restrict__ out,
                         int N, int E)
{
    // 4 waves * 2 buffers * 8 rows * 256 f32 = 64 KB static LDS
    __shared__ float smem[WAVES_PER_BLOCK][2][ROWS_PER_BATCH * D_FEAT];

    const int lane = threadIdx.x & 31;
    const int wave = threadIdx.x >> 5;
    // node is wave-uniform; readfirstlane makes that provable -> scalar control plane
    const int node = __builtin_amdgcn_readfirstlane(blockIdx.x * WAVES_PER_BLOCK + wave);
    if (node >= N) return;

    // ---- segment bounds: one merged scalar load, or fallback binary search ----
    int start, end;
    if (offs) {
        start = offs[node];
        end   = offs[node + 1];
    } else {
        int lo = 0, hi = E;
        while (lo < hi) { int mid = (lo + hi) >> 1; if (seg[mid] <  node) lo = mid + 1; else hi = mid; }
        start = lo;
        hi = E;
        while (lo < hi) { int mid = (lo + hi) >> 1; if (seg[mid] <= node) lo = mid + 1; else hi = mid; }
        end = lo;
    }
    const int deg = end - start;

    // ---- TDM gather-mode descriptor (constant parts, all SGPR) ----
    const uint64_t gaddr = (uint64_t)(uintptr_t)feat;
    const uint32_t ldsBuf0 = (uint32_t)(uintptr_t)(&smem[wave][0][0]);
    const uint32_t ldsBuf1 = (uint32_t)(uintptr_t)(&smem[wave][1][0]);

    const uint32_t g0w = (uint32_t)((gaddr >> 32) & 0x01FFFFFFu) | (2u << 30); // addr hi | type=2
    const uint32_t g0z = (uint32_t)(gaddr & 0xFFFFFFFFu);

    // Issue one tensor_load_to_lds gathering m (<=8) rows starting at edge e0.
    auto issue_gather = [&](uint32_t ldsAddr, int e0, int m) {
        // 8 uniform scalar index loads; tail clamped in-segment (extras ignored:
        // tile_dim1 = m limits valid indices)
        uint32_t ri[8];
#pragma unroll
        for (int k = 0; k < 8; ++k)
            ri[k] = (uint32_t)nbr[dmin(e0 + k, end - 1)];

        // D# group 0: count=1 | gather_index_size=32b (bit30) | gather_mode (bit31)
        su4 g0;
        g0.x = 0xC0000001u;
        g0.y = ldsAddr;          // lds_addr[31:0]
        g0.z = g0z;              // global_addr[31:0]
        g0.w = g0w;              // global_addr[56:32] | type=2

        // D# group 1
        su8 g1;
        g1.s0 = (2u << 16);                                  // data_size = 4B; wg_mask = 0
        g1.s1 = ((uint32_t)D_FEAT & 0xFFFFu) << 16;          // tensor_dim0[15:0]
        g1.s2 = (((uint32_t)N & 0xFFFFu) << 16);             // tensor_dim0[31:16]=0 | tensor_dim1[15:0]
        g1.s3 = (((uint32_t)N >> 16) & 0xFFFFu)              // tensor_dim1[31:16]
              | ((uint32_t)D_FEAT << 16);                    // tile_dim0 = 256
        g1.s4 = (uint32_t)m;                                 // tile_dim1 = #valid indices
        g1.s5 = (uint32_t)D_FEAT;                            // tensor_dim0_stride[31:0]
        g1.s6 = 0u;
        g1.s7 = 0u;

        // D# groups 2/3: 32-bit row indices 0..7
        su4 g2; g2.x = ri[0]; g2.y = ri[1]; g2.z = ri[2]; g2.w = ri[3];
        su4 g3; g3.x = ri[4]; g3.y = ri[5]; g3.z = ri[6]; g3.w = ri[7];

        asm volatile("tensor_load_to_lds %0, %1, %2, %3"
                     :
                     : "s"(g0), "s"(g1), "s"(g2), "s"(g3)
                     : "memory");
    };

    // ---- kick off first DMA batch, then load self row (overlaps DMA) ----
    const int nb = (deg + ROWS_PER_BATCH - 1) >> 3;
    if (nb > 0) issue_gather(ldsBuf0, start, dmin(ROWS_PER_BATCH, deg));

    const float* selfp = feat + (size_t)node * D_FEAT + lane * 8;
    vf4 s0 = *(const vf4*)(selfp);
    vf4 s1 = *(const vf4*)(selfp + 4);

    vf4 acc0 = (vf4)0.0f;
    vf4 acc1 = (vf4)0.0f;

    // ---- double-buffered accumulate ----
    for (int b = 0; b < nb; ++b) {
        const int e0 = start + b * ROWS_PER_BATCH;
        const int m  = dmin(ROWS_PER_BATCH, end - e0);

        if (b + 1 < nb) {
            const int e1 = e0 + ROWS_PER_BATCH;
            issue_gather((b & 1) ? ldsBuf0 : ldsBuf1, e1, dmin(ROWS_PER_BATCH, end - e1));
            __builtin_amdgcn_s_wait_tensorcnt(1);   // batch b landed (TDM in-order per wave)
        } else {
            __builtin_amdgcn_s_wait_tensorcnt(0);   // last batch landed
        }
        asm volatile("" ::: "memory");              // keep LDS reads below the wait

        const float* rb = &smem[wave][b & 1][0] + lane * 8;
#pragma unroll 4
        for (int r = 0; r < m; ++r) {
            vf4 x0 = *(const vf4*)(rb + r * D_FEAT);
            vf4 x1 = *(const vf4*)(rb + r * D_FEAT + 4);
            acc0 += x0;
            acc1 += x1;
        }
    }

    // ---- self loop + mean + store ----
    const float inv = 1.0f / (float)(deg + 1);
    vf4 o0 = (acc0 + s0) * inv;
    vf4 o1 = (acc1 + s1) * inv;

    float* op = out + (size_t)node * D_FEAT + lane * 8;
    *(vf4*)(op)     = o0;
    *(vf4*)(op + 4) = o1;
}

extern "C" void kernel_launch(void* const* d_in, const int* in_sizes, int n_in,
                              void* d_out, int out_size, void* d_ws, size_t ws_size,
                              hipStream_t stream) {
    const float* feat = (const float*)d_in[0];
    const int*   nbr  = (const int*)d_in[1];
    const int*   seg  = (const int*)d_in[2];
    float*       out  = (float*)d_out;

    const int N = in_sizes[0] / D_FEAT;   // 50000 nodes
    const int E = in_sizes[1];            // 1.6M edges

    int* offs = nullptr;
    if (d_ws && ws_size >= (size_t)(N + 1) * sizeof(int)) {
        offs = (int*)d_ws;
        segment_bounds_kernel<<<(E + 255) / 256, 256, 0, stream>>>(seg, offs, N, E);
    }

    const int grid = (N + WAVES_PER_BLOCK - 1) / WAVES_PER_BLOCK;
    gcn_aggregate_tdm_kernel<<<grid, THREADS_PER_BLOCK, 0, stream>>>(feat, nbr, seg, offs,
                                                                     out, N, E);
}